// LSTM_90142773608947
// MI455X (gfx1250) — compile-verified
//
#include <hip/hip_runtime.h>
#include <hip/hip_bf16.h>

// LSTM tagger fused kernel for gfx1250 (MI455X).
// B=1024, T=512, E=16, H=32, 4H=128, NUM_TAGS=11.
// 64 workgroups x 256 threads (8 wave32 waves). Each WG owns 16 batch rows
// for the full sequence; wave w owns gate columns [16w, 16w+16).
// All matmuls are V_WMMA_F32_16X16X4_F32 with weights held in registers.
// Round-3 change: two-stage software pipeline for the token->embedding
// gather chain (token index loaded 2 steps ahead, embedding row 1 step
// ahead) so no global load is waited on in the iteration that issues it.

typedef __attribute__((ext_vector_type(2))) float v2f;
typedef __attribute__((ext_vector_type(8))) float v8f;

namespace {
constexpr int Bc = 1024;
constexpr int Tc = 512;
constexpr int Ec = 16;    // embed
constexpr int Hc = 32;    // hidden
constexpr int NT = 11;    // num tags
}

__device__ __forceinline__ float fast_rcp(float x) {
  return __builtin_amdgcn_rcpf(x);      // single v_rcp_f32
}
__device__ __forceinline__ float sigm_f(float x) {
  return fast_rcp(1.0f + __expf(-x));
}
__device__ __forceinline__ float tanh_f(float x) {
  // tanh(x) = 2*sigmoid(2x) - 1
  return 2.0f * fast_rcp(1.0f + __expf(-2.0f * x)) - 1.0f;
}

__global__ __launch_bounds__(256) void lstm_fused_kernel(
    const int* __restrict__ x,       // [B, T]
    const float* __restrict__ emb,   // [VOCAB, 16]
    const float* __restrict__ W_ih,  // [128, 16]
    const float* __restrict__ W_hh,  // [128, 32]
    const float* __restrict__ b_ih,  // [128]
    const float* __restrict__ b_hh,  // [128]
    const float* __restrict__ W_fc,  // [11, 32]
    const float* __restrict__ b_fc,  // [11]
    float* __restrict__ out)         // [B, T, 11]
{
  const int tid   = threadIdx.x;        // 0..255
  const int wave  = tid >> 5;           // 0..7
  const int lane  = tid & 31;           // 0..31
  const int lhalf = lane >> 4;          // 0 or 1
  const int lm    = lane & 15;          // 0..15
  const int b0    = blockIdx.x * 16;    // batch tile base

  // LDS state (padded strides to reduce bank conflicts)
  __shared__ float sh_h[16][36];        // h_t  [row][hid]
  __shared__ float sh_g[16][132];       // gate pre-activations [row][gate]

  // ---- preload this wave's weight slices as WMMA B operands ----
  // B element for (N-col n, K-pair k0): b = {W[n][k0], W[n][k0+1]},
  // n = 16*wave + lm, k0 = 2*lhalf + 4*kstep.
  const int n = wave * 16 + lm;

  v2f wih[4];
#pragma unroll
  for (int kk = 0; kk < 4; ++kk) {
    const int k0 = 4 * kk + 2 * lhalf;
    wih[kk].x = W_ih[n * Ec + k0];
    wih[kk].y = W_ih[n * Ec + k0 + 1];
  }
  v2f whh[8];
#pragma unroll
  for (int kk = 0; kk < 8; ++kk) {
    const int k0 = 4 * kk + 2 * lhalf;
    whh[kk].x = W_hh[n * Hc + k0];
    whh[kk].y = W_hh[n * Hc + k0 + 1];
  }
  const float bs = b_ih[n] + b_hh[n];   // combined bias for this gate column

  // wave 0 additionally holds W_fc^T tiles (cols >= 11 zero-padded)
  v2f wfc[8];
  float bfcn = 0.0f;
#pragma unroll
  for (int kk = 0; kk < 8; ++kk) { wfc[kk].x = 0.0f; wfc[kk].y = 0.0f; }
  if (wave == 0 && lm < NT) {
#pragma unroll
    for (int kk = 0; kk < 8; ++kk) {
      const int k0 = 4 * kk + 2 * lhalf;
      wfc[kk].x = W_fc[lm * Hc + k0];
      wfc[kk].y = W_fc[lm * Hc + k0 + 1];
    }
    bfcn = b_fc[lm];
  }

  // ---- init h = 0 in LDS, c = 0 in registers ----
  for (int i = tid; i < 16 * 36; i += 256) ((float*)sh_h)[i] = 0.0f;
  float creg[2] = {0.0f, 0.0f};
  __syncthreads();

  const int xrow = (b0 + lm) * Tc;      // token-index row for batch row lm

  // ---- two-stage pipeline prologue ----
  // ea       : embedding rows for step t   (ready)
  // idx_next : token index for step t+1    (ready)
  v2f ea[4];
  int idx_next;
  {
    const int idx0 = x[xrow];
    const float* er0 = emb + (long)idx0 * Ec;
#pragma unroll
    for (int kk = 0; kk < 4; ++kk) ea[kk] = *(const v2f*)(er0 + 4 * kk + 2 * lhalf);
    idx_next = x[xrow + 1];
  }

  for (int t = 0; t < Tc; ++t) {
    // branchless lookahead prefetch of token indices (global_prefetch_b8)
    {
      const int tp = (t + 16 < Tc) ? (t + 16) : (Tc - 1);
      __builtin_prefetch(x + xrow + tp, 0, 3);
    }

    // bias-initialized accumulator chain + zero-initialized second chain
    v8f acc0, acc1;
#pragma unroll
    for (int i = 0; i < 8; ++i) { acc0[i] = bs; acc1[i] = 0.0f; }

    // input projection: emb(16x16) @ W_ih^T  (uses pipelined ea regs)
#pragma unroll
    for (int kk = 0; kk < 4; ++kk) {
      acc0 = __builtin_amdgcn_wmma_f32_16x16x4_f32(
          false, ea[kk], false, wih[kk], (short)0, acc0, false, false);
    }

    // Stage 1: gather embedding rows for t+1 using idx_next, which was
    // loaded a full iteration ago (its wait is essentially free now).
    v2f ean[4];
    {
      const float* ern = emb + (long)idx_next * Ec;
#pragma unroll
      for (int kk = 0; kk < 4; ++kk) ean[kk] = *(const v2f*)(ern + 4 * kk + 2 * lhalf);
    }
    // Stage 2: issue token load for t+2; no consumer until next iteration.
    {
      const int t2 = (t + 2 < Tc) ? (t + 2) : (Tc - 1);
      idx_next = x[xrow + t2];
    }

    // hidden projection: h(16x32) @ W_hh^T, two independent 4-deep chains
#pragma unroll
    for (int kk = 0; kk < 4; ++kk) {
      const int k0a = 4 * kk + 2 * lhalf;
      const int k0b = 16 + 4 * kk + 2 * lhalf;
      v2f a0 = *(const v2f*)(&sh_h[lm][k0a]);
      v2f a1 = *(const v2f*)(&sh_h[lm][k0b]);
      acc0 = __builtin_amdgcn_wmma_f32_16x16x4_f32(
          false, a0, false, whh[kk], (short)0, acc0, false, false);
      acc1 = __builtin_amdgcn_wmma_f32_16x16x4_f32(
          false, a1, false, whh[kk + 4], (short)0, acc1, false, false);
    }

    // D element i -> (row = i + 8*lhalf, col = n)
#pragma unroll
    for (int i = 0; i < 8; ++i) sh_g[i + 8 * lhalf][n] = acc0[i] + acc1[i];
    __syncthreads();

    // ---- elementwise LSTM cell update: 512 (row,hid) pairs / 256 lanes ----
#pragma unroll
    for (int j = 0; j < 2; ++j) {
      const int e   = tid + 256 * j;    // 0..511
      const int row = e & 15;
      const int hid = e >> 4;           // 0..31
      const float gi = sh_g[row][hid];
      const float gf = sh_g[row][32 + hid];
      const float gg = sh_g[row][64 + hid];
      const float go = sh_g[row][96 + hid];
      const float c_new = sigm_f(gf) * creg[j] + sigm_f(gi) * tanh_f(gg);
      creg[j] = c_new;
      sh_h[row][hid] = sigm_f(go) * tanh_f(c_new);
    }
    __syncthreads();

    // ---- wave 0: logits tile = h(16x32) @ W_fc^T + b_fc, store 11 cols ----
    if (wave == 0) {
      v8f lacc0, lacc1;
#pragma unroll
      for (int i = 0; i < 8; ++i) { lacc0[i] = bfcn; lacc1[i] = 0.0f; }
#pragma unroll
      for (int kk = 0; kk < 4; ++kk) {
        const int k0a = 4 * kk + 2 * lhalf;
        const int k0b = 16 + 4 * kk + 2 * lhalf;
        v2f a0 = *(const v2f*)(&sh_h[lm][k0a]);
        v2f a1 = *(const v2f*)(&sh_h[lm][k0b]);
        lacc0 = __builtin_amdgcn_wmma_f32_16x16x4_f32(
            false, a0, false, wfc[kk], (short)0, lacc0, false, false);
        lacc1 = __builtin_amdgcn_wmma_f32_16x16x4_f32(
            false, a1, false, wfc[kk + 4], (short)0, lacc1, false, false);
      }
      if (lm < NT) {
#pragma unroll
        for (int i = 0; i < 8; ++i) {
          const int row = i + 8 * lhalf;
          out[((long)(b0 + row) * Tc + t) * NT + lm] = lacc0[i] + lacc1[i];
        }
      }
    }

    // rotate pipeline registers
#pragma unroll
    for (int kk = 0; kk < 4; ++kk) ea[kk] = ean[kk];
  }
}

extern "C" void kernel_launch(void* const* d_in, const int* in_sizes, int n_in,
                              void* d_out, int out_size, void* d_ws, size_t ws_size,
                              hipStream_t stream) {
  const int*   x    = (const int*)d_in[0];
  const float* emb  = (const float*)d_in[1];
  const float* W_ih = (const float*)d_in[2];
  const float* W_hh = (const float*)d_in[3];
  const float* b_ih = (const float*)d_in[4];
  const float* b_hh = (const float*)d_in[5];
  const float* W_fc = (const float*)d_in[6];
  const float* b_fc = (const float*)d_in[7];
  float* out = (float*)d_out;

  lstm_fused_kernel<<<dim3(Bc / 16), dim3(256), 0, stream>>>(
      x, emb, W_ih, W_hh, b_ih, b_hh, W_fc, b_fc, out);
}